// PhotonicInterferenceLayer_76338748719654
// MI455X (gfx1250) — compile-verified
//
#include <hip/hip_runtime.h>
#include <math.h>

typedef float v2f __attribute__((ext_vector_type(2)));
typedef float v8f __attribute__((ext_vector_type(8)));

#define CDIM 32          // 2*EMBED interleaved (r,i)
#define RELMAX 64
#define CHUNK 4096
#define INV_SQRT_D 0.5f  // 1/sqrt(HEAD=4)

// ---------------------------------------------------------------------------
// Projection: kb[l][32] = z_past[l] @ k_w^T + k_b + rel_bias[idx(l)]
//             v [l][32] = z_past[l] @ v_w^T + v_b
// One wave per 16-row tile, WMMA f32 16x16x4, K=32 in 8 chunks, N=32 in 2 tiles.
// ---------------------------------------------------------------------------
__global__ void proj_kernel(const float* __restrict__ zp,
                            const float* __restrict__ kw, const float* __restrict__ kbias,
                            const float* __restrict__ vw, const float* __restrict__ vbias,
                            const float* __restrict__ relb, const int* __restrict__ posp,
                            float* __restrict__ kbout, float* __restrict__ vout, int Lp)
{
    const int lane = threadIdx.x & 31;
    const int wave = threadIdx.x >> 5;
    const long base = (long)blockIdx.x * 128 + (long)wave * 16;
    if (base >= Lp) return;                      // uniform per wave -> EXEC stays full
    const int nl   = lane & 15;                  // M (for A) / N (for B,C)
    const int koff = (lane >> 4) * 2;            // K split across lane halves
    long r = base + nl;
    if (r >= Lp) r = (long)Lp - 1;               // clamp loads; stores guarded below

    v8f ck0, ck1, cv0, cv1;
    const float bk0 = kbias[nl], bk1 = kbias[nl + 16];
    const float bv0 = vbias[nl], bv1 = vbias[nl + 16];
#pragma unroll
    for (int j = 0; j < 8; ++j) { ck0[j] = bk0; ck1[j] = bk1; cv0[j] = bv0; cv1[j] = bv1; }

    const float* arow = zp + r * CDIM + koff;
#pragma unroll
    for (int kc = 0; kc < 8; ++kc) {
        v2f a = *(const v2f*)(arow + kc * 4);                       // A: Z[r, 4kc+koff..+1]
        v2f wk0 = *(const v2f*)(kw + (size_t)nl * CDIM + kc * 4 + koff);        // B = W^T tile
        v2f wk1 = *(const v2f*)(kw + (size_t)(nl + 16) * CDIM + kc * 4 + koff);
        v2f wv0 = *(const v2f*)(vw + (size_t)nl * CDIM + kc * 4 + koff);
        v2f wv1 = *(const v2f*)(vw + (size_t)(nl + 16) * CDIM + kc * 4 + koff);
        ck0 = __builtin_amdgcn_wmma_f32_16x16x4_f32(false, a, false, wk0, (short)0, ck0, false, false);
        ck1 = __builtin_amdgcn_wmma_f32_16x16x4_f32(false, a, false, wk1, (short)0, ck1, false, false);
        cv0 = __builtin_amdgcn_wmma_f32_16x16x4_f32(false, a, false, wv0, (short)0, cv0, false, false);
        cv1 = __builtin_amdgcn_wmma_f32_16x16x4_f32(false, a, false, wv1, (short)0, cv1, false, false);
    }

    const int roff = posp[0] - Lp + RELMAX;      // rel_bias row offset (clipped per row)
#pragma unroll
    for (int j = 0; j < 8; ++j) {
        const int m = (lane < 16) ? j : j + 8;   // C layout: lanes 0-15 -> M=j, 16-31 -> M=j+8
        const long row = base + m;
        if (row < Lp) {
            int idx = (int)row + roff;
            idx = idx < 0 ? 0 : (idx > 2 * RELMAX ? 2 * RELMAX : idx);
            kbout[row * CDIM + nl]      = ck0[j] + relb[(size_t)idx * CDIM + nl];
            kbout[row * CDIM + nl + 16] = ck1[j] + relb[(size_t)idx * CDIM + nl + 16];
            vout[row * CDIM + nl]      = cv0[j];
            vout[row * CDIM + nl + 16] = cv1[j];
        }
    }
}

// ---------------------------------------------------------------------------
// Init: z0 = candidate, q0 = q_w @ z0 + q_b
// ---------------------------------------------------------------------------
__global__ void init_kernel(const float* __restrict__ cand,
                            const float* __restrict__ qw, const float* __restrict__ qb,
                            float* __restrict__ zbuf, float* __restrict__ qbuf)
{
    __shared__ float zc[32];
    const int tid = threadIdx.x;        // 32 threads
    zc[tid] = cand[tid];
    zbuf[tid] = cand[tid];
    __syncthreads();
    float acc = qb[tid];
#pragma unroll
    for (int j = 0; j < 32; ++j) acc += qw[tid * 32 + j] * zc[j];
    qbuf[tid] = acc;
}

// ---------------------------------------------------------------------------
// Attention partials: online softmax per (block, head) over a CHUNK of rows.
// tid&3 = head, 64 threads per head stride the rows.
// ---------------------------------------------------------------------------
__global__ void attn_kernel(const float* __restrict__ kbmat, const float* __restrict__ vmat,
                            const float* __restrict__ q, float* __restrict__ part, int Lp)
{
    __shared__ float red[256][10];
    const int tid = threadIdx.x;
    const int h = tid & 3;
    const long start = (long)blockIdx.x * CHUNK;
    long end = start + CHUNK; if (end > Lp) end = Lp;

    float qv[8];
#pragma unroll
    for (int j = 0; j < 8; ++j) qv[j] = q[h * 8 + j];

    float m = -INFINITY, s = 0.f;
    float vs[8];
#pragma unroll
    for (int j = 0; j < 8; ++j) vs[j] = 0.f;

    for (long l = start + (tid >> 2); l < end; l += 64) {
        const float4* k4 = (const float4*)(kbmat + l * CDIM + h * 8);
        const float4* v4 = (const float4*)(vmat  + l * CDIM + h * 8);
        float4 ka = k4[0], kb = k4[1];
        float dot = qv[0]*ka.x + qv[1]*ka.y + qv[2]*ka.z + qv[3]*ka.w
                  + qv[4]*kb.x + qv[5]*kb.y + qv[6]*kb.z + qv[7]*kb.w;
        float x = dot * INV_SQRT_D;
        float mn = fmaxf(m, x);
        float eo = __expf(m - mn);
        float en = __expf(x - mn);
        s = s * eo + en;
        float4 va = v4[0], vb = v4[1];
        vs[0] = vs[0]*eo + en*va.x;  vs[1] = vs[1]*eo + en*va.y;
        vs[2] = vs[2]*eo + en*va.z;  vs[3] = vs[3]*eo + en*va.w;
        vs[4] = vs[4]*eo + en*vb.x;  vs[5] = vs[5]*eo + en*vb.y;
        vs[6] = vs[6]*eo + en*vb.z;  vs[7] = vs[7]*eo + en*vb.w;
        m = mn;
    }

    red[tid][0] = m; red[tid][1] = s;
#pragma unroll
    for (int j = 0; j < 8; ++j) red[tid][2 + j] = vs[j];
    __syncthreads();

    if (tid < 4) {
        float M = -INFINITY, S = 0.f, VS[8];
#pragma unroll
        for (int j = 0; j < 8; ++j) VS[j] = 0.f;
        for (int t = tid; t < 256; t += 4) {
            float m2 = red[t][0], s2 = red[t][1];
            if (s2 > 0.f) {
                float mn = fmaxf(M, m2);
                float e1 = (S > 0.f) ? __expf(M - mn) : 0.f;
                float e2 = __expf(m2 - mn);
                S = S * e1 + s2 * e2;
                for (int j = 0; j < 8; ++j) VS[j] = VS[j] * e1 + red[t][2 + j] * e2;
                M = mn;
            }
        }
        float* p = part + ((long)blockIdx.x * 4 + tid) * 10;
        p[0] = M; p[1] = S;
        for (int j = 0; j < 8; ++j) p[2 + j] = VS[j];
    }
}

// ---------------------------------------------------------------------------
// Reduce partials + o-proj + mish + layernorm + next q-proj (one tiny block).
// ---------------------------------------------------------------------------
__global__ void update_kernel(const float* __restrict__ part, int NB,
                              const float* __restrict__ ow, const float* __restrict__ ob,
                              const float* __restrict__ qw, const float* __restrict__ qb,
                              const float* __restrict__ coupling, const float* __restrict__ nscale,
                              float* __restrict__ zbuf, float* __restrict__ qbuf,
                              float* __restrict__ dout, int last)
{
    __shared__ float outr[32], zm[32], zn[32];
    const int tid = threadIdx.x;   // 64 threads

    if (tid < 4) {
        float M = -INFINITY, S = 0.f, VS[8];
#pragma unroll
        for (int j = 0; j < 8; ++j) VS[j] = 0.f;
        for (int b = 0; b < NB; ++b) {
            const float* p = part + ((long)b * 4 + tid) * 10;
            float m2 = p[0], s2 = p[1];
            if (s2 > 0.f) {
                float mn = fmaxf(M, m2);
                float e1 = (S > 0.f) ? __expf(M - mn) : 0.f;
                float e2 = __expf(m2 - mn);
                S = S * e1 + s2 * e2;
                for (int j = 0; j < 8; ++j) VS[j] = VS[j] * e1 + p[2 + j] * e2;
                M = mn;
            }
        }
        float inv = 1.f / S;
        for (int j = 0; j < 8; ++j) outr[tid * 8 + j] = VS[j] * inv;
    }
    __syncthreads();

    if (tid < 32) {
        float acc = ob[tid];
#pragma unroll
        for (int j = 0; j < 32; ++j) acc += ow[tid * 32 + j] * outr[j];
        float zi = zbuf[tid] + coupling[0] * acc;
        float sp = (zi > 20.f) ? zi : log1pf(__expf(zi));   // softplus
        zm[tid] = zi * tanhf(sp);                           // mish
    }
    __syncthreads();

    if (tid < 32) {
        const int p = tid & 1;          // 0 = real lane, 1 = imag lane
        float mean = 0.f;
        for (int t = 0; t < 16; ++t) mean += zm[2 * t + p];
        mean *= (1.f / 16.f);
        float var = 0.f;
        for (int t = 0; t < 16; ++t) { float d = zm[2 * t + p] - mean; var += d * d; }
        var *= (1.f / 16.f);
        float znv = (zm[tid] - mean) * rsqrtf(var + 1e-5f) * nscale[0];
        zn[tid] = znv;
        zbuf[tid] = znv;
        if (last) dout[tid] = znv;
    }
    __syncthreads();

    if (tid < 32) {
        float acc = qb[tid];
#pragma unroll
        for (int j = 0; j < 32; ++j) acc += qw[tid * 32 + j] * zn[j];
        qbuf[tid] = acc;
    }
}

// ---------------------------------------------------------------------------
extern "C" void kernel_launch(void* const* d_in, const int* in_sizes, int n_in,
                              void* d_out, int out_size, void* d_ws, size_t ws_size,
                              hipStream_t stream)
{
    const float* cand = (const float*)d_in[0];
    const float* zp   = (const float*)d_in[1];
    const float* qw   = (const float*)d_in[2];
    const float* qb   = (const float*)d_in[3];
    const float* kw   = (const float*)d_in[4];
    const float* kbb  = (const float*)d_in[5];
    const float* vw   = (const float*)d_in[6];
    const float* vbb  = (const float*)d_in[7];
    const float* ow   = (const float*)d_in[8];
    const float* obb  = (const float*)d_in[9];
    const float* relb = (const float*)d_in[10];
    const float* coup = (const float*)d_in[11];
    const float* nsc  = (const float*)d_in[12];
    const int*   posp = (const int*)d_in[13];

    const int Lp = in_sizes[1] / CDIM;
    float* ws   = (float*)d_ws;
    float* kbuf = ws;                                  // [Lp*32]
    float* vbuf = ws + (size_t)Lp * CDIM;              // [Lp*32]
    float* zbuf = vbuf + (size_t)Lp * CDIM;            // [32]
    float* qbuf = zbuf + 32;                           // [32]
    float* part = qbuf + 32;                           // [NB*4*10]

    const int NB = (Lp + CHUNK - 1) / CHUNK;
    const int PB = (Lp + 127) / 128;

    proj_kernel<<<PB, 256, 0, stream>>>(zp, kw, kbb, vw, vbb, relb, posp, kbuf, vbuf, Lp);
    init_kernel<<<1, 32, 0, stream>>>(cand, qw, qb, zbuf, qbuf);

    for (int step = 0; step < 8; ++step) {
        attn_kernel<<<NB, 256, 0, stream>>>(kbuf, vbuf, qbuf, part, Lp);
        update_kernel<<<1, 64, 0, stream>>>(part, NB, ow, obb, qw, qb, coup, nsc,
                                            zbuf, qbuf, (float*)d_out, step == 7 ? 1 : 0);
    }
}